// RoIAlign_77060303225121
// MI455X (gfx1250) — compile-verified
//
#include <hip/hip_runtime.h>
#include <hip/hip_bf16.h>
#include <stdint.h>

// ROIAlign for MI455X (gfx1250, wave32).
// x: [B=2, C=256, H=200, W=200] f32; rois: [K, 5] f32; out: [K, 256, 7, 7] f32.
// Bandwidth-bound (input fits 192MB L2): gather pipeline + async-to-LDS staging.

#define POOLED 7
#define NBINS  49            // 7*7
#define NSAMP  14            // POOLED * sampling_ratio(2)
#define CCH    256
#define HH     200
#define WW     200
#define SP_SCALE 0.25f

struct Samp { int lo; int hi; float w0; float w1; };  // 16B -> ds_load_b128

#if defined(__gfx1250__) && __has_builtin(__builtin_amdgcn_global_load_async_to_lds_b32)
#define HAVE_ASYNC_LDS 1
#else
#define HAVE_ASYNC_LDS 0
#endif

typedef __attribute__((address_space(1))) int gint_t;
typedef __attribute__((address_space(3))) int lint_t;

__device__ __forceinline__ void wait_async0() {
#if __has_builtin(__builtin_amdgcn_s_wait_asynccnt)
  __builtin_amdgcn_s_wait_asynccnt(0);
#else
  asm volatile("s_wait_asynccnt 0x0" ::: "memory");
#endif
}

__global__ __launch_bounds__(256) void roi_align_main(
    const float* __restrict__ x, const float* __restrict__ rois,
    float* __restrict__ out, int K) {
  __shared__ float s_roi[8];
  __shared__ Samp  s_y[NSAMP];
  __shared__ Samp  s_x[NSAMP];
  __shared__ int   s_bidx;

  const int blk   = blockIdx.x;
  const int k     = blk / NBINS;        // ROI index
  const int chunk = blk - k * NBINS;    // which 256-element chunk of this ROI
  const int t     = threadIdx.x;

  // ---- wave 0: stage ROI row into LDS (CDNA5 async-to-LDS path) + build tables
  if (t < 32) {
    if (t < 5) {
#if HAVE_ASYNC_LDS
      __builtin_amdgcn_global_load_async_to_lds_b32(
          (gint_t*)(rois + (size_t)k * 5 + t),
          (lint_t*)(&s_roi[t]),
          /*offset=*/0, /*cpol=*/0);
#else
      s_roi[t] = rois[(size_t)k * 5 + t];
#endif
    }
#if HAVE_ASYNC_LDS
    wait_async0();   // scalar: whole wave 0 waits before reading s_roi
#endif
    if (t < 2 * NSAMP) {
      const bool  isY   = (t < NSAMP);
      const int   jj    = isY ? t : (t - NSAMP);
      const float start = s_roi[isY ? 2 : 1] * SP_SCALE - 0.5f;
      const float end   = s_roi[isY ? 4 : 3] * SP_SCALE - 0.5f;
      const float bin   = (end - start) * (1.0f / (float)POOLED);
      const int   p     = jj >> 1;
      const float s     = ((float)(jj & 1) + 0.5f) * 0.5f;   // (i+0.5)/S, S=2
      const float coord = start + ((float)p + s) * bin;
      const float limit = (float)(isY ? HH : WW);             // both 200
      const float valid = (coord >= -1.0f && coord <= limit) ? 1.0f : 0.0f;
      float c = fmaxf(coord, 0.0f);
      int low0 = (int)floorf(c);
      int lo, hi;
      if (low0 >= (int)limit - 1) {          // cap: lo = hi = limit-1, l = 0
        lo = (int)limit - 1; hi = lo; c = (float)lo;
      } else {
        lo = low0; hi = low0 + 1;
      }
      const float l = c - (float)lo;
      const float h = 1.0f - l;
      Samp sm; sm.lo = lo; sm.hi = hi; sm.w0 = h * valid; sm.w1 = l * valid;
      if (isY) s_y[jj] = sm; else s_x[jj] = sm;
      if (t == 0) s_bidx = (int)s_roi[0];
    }
  }
  __syncthreads();

  // ---- per-thread output element: idx = c*49 + ph*7 + pw  (pw fastest lane-wise)
  const int idx = chunk * 256 + t;            // 0 .. 12543
  const int c   = idx / NBINS;
  const int bin = idx - c * NBINS;
  const int ph  = bin / POOLED;
  const int pw  = bin - ph * POOLED;

  const Samp y0 = s_y[2 * ph], y1 = s_y[2 * ph + 1];
  const Samp x0 = s_x[2 * pw], x1 = s_x[2 * pw + 1];

  const float* __restrict__ base =
      x + ((size_t)s_bidx * CCH + (size_t)c) * (size_t)(HH * WW);

  float acc = 0.0f;
#pragma unroll
  for (int sy = 0; sy < 2; ++sy) {
    const Samp ys = sy ? y1 : y0;
    const float* __restrict__ r0 = base + (size_t)ys.lo * WW;
    const float* __restrict__ r1 = base + (size_t)ys.hi * WW;
#pragma unroll
    for (int sx = 0; sx < 2; ++sx) {
      const Samp xs = sx ? x1 : x0;
      const float vt = xs.w0 * r0[xs.lo] + xs.w1 * r0[xs.hi];
      const float vb = xs.w0 * r1[xs.lo] + xs.w1 * r1[xs.hi];
      acc += ys.w0 * vt + ys.w1 * vb;
    }
  }

  const size_t o = (size_t)k * (size_t)(CCH * NBINS) + (size_t)idx;
  __builtin_nontemporal_store(acc * 0.25f, &out[o]);   // avg over 2x2 samples; NT: keep L2 for x
}

extern "C" void kernel_launch(void* const* d_in, const int* in_sizes, int n_in,
                              void* d_out, int out_size, void* d_ws, size_t ws_size,
                              hipStream_t stream) {
  (void)n_in; (void)out_size; (void)d_ws; (void)ws_size;
  const float* x    = (const float*)d_in[0];
  const float* rois = (const float*)d_in[1];
  float* out        = (float*)d_out;
  const int K = in_sizes[1] / 5;               // 1000
  dim3 grid((unsigned)(K * NBINS));            // one block per (roi, 256-elem chunk)
  roi_align_main<<<grid, 256, 0, stream>>>(x, rois, out, K);
}